// VolumeRenderer_14336600834233
// MI455X (gfx1250) — compile-verified
//
#include <hip/hip_runtime.h>
#include <hip/hip_bf16.h>
#include <stdint.h>

// Volume rendering compositing for MI455X (gfx1250), wave32.
// Memory-bound (AI ~0.5 flop/byte): design around 23.3 TB/s HBM streaming.
// One wave per ray: lane i owns samples [4i, 4i+3]. Cumulative transmittance
// via wave32 shfl_up prefix-product scan; per-ray tiles (alpha 512B + rgb
// 1536B) are double-buffered in LDS via gfx1250 async global->LDS b128 copies
// tracked by ASYNCcnt.

#define SAMPLES        128
#define WAVES_PER_BLK  8
#define RAYS_PER_WAVE  4
#define TILE_FLOATS    512            // 128 alpha + 384 rgb
#define WAVE_LDS       (2 * TILE_FLOATS)
#define EPSV           1e-10f

#if defined(__has_builtin)
#  if __has_builtin(__builtin_amdgcn_global_load_async_to_lds_b128)
#    define HAVE_ASYNC_B128 1
#  endif
#  if __has_builtin(__builtin_amdgcn_s_wait_asynccnt)
#    define HAVE_WAIT_ASYNC 1
#  endif
#endif

#if defined(__HIP_DEVICE_COMPILE__)
typedef int v4i __attribute__((ext_vector_type(4)));
typedef __attribute__((address_space(1))) v4i* gptr_v4i;   // global (AS1)
typedef __attribute__((address_space(3))) v4i* lptr_v4i;   // LDS (AS3)
#endif

__device__ __forceinline__ void async_copy_b128(const float* gsrc, float* ldst) {
#if defined(HAVE_ASYNC_B128) && defined(__HIP_DEVICE_COMPILE__)
  __builtin_amdgcn_global_load_async_to_lds_b128(
      (gptr_v4i)(const_cast<float*>(gsrc)),   // addrspacecast generic -> AS1
      (lptr_v4i)(ldst),                       // addrspacecast generic -> AS3
      /*offset=*/0, /*cpol=*/0);
#else
  // Fallback: synchronous copy through VGPRs (global_load_b128 + ds_store_b128).
  *(float4*)ldst = *(const float4*)gsrc;
#endif
}

template <int N>
__device__ __forceinline__ void wait_asynccnt() {
#if defined(HAVE_WAIT_ASYNC)
  __builtin_amdgcn_s_wait_asynccnt(N);
#elif defined(HAVE_ASYNC_B128)
  asm volatile("s_wait_asynccnt %0" :: "i"(N) : "memory");
#else
  // Synchronous fallback path: nothing outstanding.
#endif
}

__global__ __launch_bounds__(WAVES_PER_BLK * 32)
void volrender_kernel(const float* __restrict__ alpha,
                      const float* __restrict__ rgbs,
                      float* __restrict__ out,
                      int n_rays) {
  __shared__ float lds[WAVES_PER_BLK * WAVE_LDS];   // 32 KB

  const int lane = threadIdx.x & 31;
  const int wave = threadIdx.x >> 5;
  float* wbuf = &lds[wave * WAVE_LDS];

  const int ray0 = ((int)blockIdx.x * WAVES_PER_BLK + wave) * RAYS_PER_WAVE;

  // Issue the 4 async b128 copies for one ray tile (per-wave private buffer,
  // so only ASYNCcnt ordering is needed — no barriers).
  auto issue_tile = [&](int ray, int buf) {
    const float* ga = alpha + (size_t)ray * SAMPLES;
    const float* gr = rgbs  + (size_t)ray * SAMPLES * 3;
    float* la = wbuf + buf * TILE_FLOATS;
    float* lr = la + SAMPLES;
    async_copy_b128(ga + lane * 4,        la + lane * 4);        // 512 B alpha
    async_copy_b128(gr + lane * 4,        lr + lane * 4);        // rgb 0..511 B
    async_copy_b128(gr + 128 + lane * 4,  lr + 128 + lane * 4);  // rgb 512..1023
    async_copy_b128(gr + 256 + lane * 4,  lr + 256 + lane * 4);  // rgb 1024..1535
  };

  if (ray0 < n_rays) issue_tile(ray0, 0);

  for (int i = 0; i < RAYS_PER_WAVE; ++i) {
    const int ray = ray0 + i;
    if (ray >= n_rays) break;

    const bool pf = (i + 1 < RAYS_PER_WAVE) && (ray + 1 < n_rays);
    if (pf) {
      issue_tile(ray + 1, (i + 1) & 1);
      wait_asynccnt<4>();   // current tile done; prefetch (4 copies) in flight
    } else {
      wait_asynccnt<0>();
    }

    const float* la = wbuf + (i & 1) * TILE_FLOATS;
    const float* lr = la + SAMPLES;

    // ds_load_b128 x4 per lane
    const float4 a  = *(const float4*)(la + lane * 4);
    const float4 g0 = *(const float4*)(lr + lane * 12);
    const float4 g1 = *(const float4*)(lr + lane * 12 + 4);
    const float4 g2 = *(const float4*)(lr + lane * 12 + 8);

    // Per-lane transmittance factors and local product.
    const float f0 = 1.0f - a.x + EPSV;
    const float f1 = 1.0f - a.y + EPSV;
    const float f2 = 1.0f - a.z + EPSV;
    const float f3 = 1.0f - a.w + EPSV;
    float incl = (f0 * f1) * (f2 * f3);

    // Wave32 inclusive prefix product (5 shfl_up steps).
    #pragma unroll
    for (int off = 1; off < 32; off <<= 1) {
      const float u = __shfl_up(incl, off, 32);
      if (lane >= off) incl *= u;
    }
    // Exclusive prefix = transmittance entering this lane's first sample.
    float t = __shfl_up(incl, 1, 32);
    if (lane == 0) t = 1.0f;

    // weights[s] = alpha[s] * prod_{k<s}(1-alpha[k]+eps)  (shift-by-1 semantics)
    float s0 = 0.0f, s1 = 0.0f, s2 = 0.0f, w;
    w = t * a.x; s0 += w * g0.x; s1 += w * g0.y; s2 += w * g0.z; t *= f0;
    w = t * a.y; s0 += w * g0.w; s1 += w * g1.x; s2 += w * g1.y; t *= f1;
    w = t * a.z; s0 += w * g1.z; s1 += w * g1.w; s2 += w * g2.x; t *= f2;
    w = t * a.w; s0 += w * g2.y; s1 += w * g2.z; s2 += w * g2.w; t *= f3;

    // Full-wave butterfly reduction per channel.
    #pragma unroll
    for (int off = 16; off > 0; off >>= 1) {
      s0 += __shfl_xor(s0, off, 32);
      s1 += __shfl_xor(s1, off, 32);
      s2 += __shfl_xor(s2, off, 32);
    }

    if (lane < 3) {
      const float v = (lane == 0) ? s0 : (lane == 1) ? s1 : s2;
      out[(size_t)ray * 3 + lane] = v;
    }
  }
}

extern "C" void kernel_launch(void* const* d_in, const int* in_sizes, int n_in,
                              void* d_out, int out_size, void* d_ws, size_t ws_size,
                              hipStream_t stream) {
  (void)n_in; (void)out_size; (void)d_ws; (void)ws_size;
  const float* alpha = (const float*)d_in[0];
  const float* rgbs  = (const float*)d_in[1];
  float* out = (float*)d_out;

  const int n_rays = in_sizes[0] / SAMPLES;
  const int rays_per_block = WAVES_PER_BLK * RAYS_PER_WAVE;
  const int grid = (n_rays + rays_per_block - 1) / rays_per_block;

  hipLaunchKernelGGL(volrender_kernel, dim3(grid), dim3(WAVES_PER_BLK * 32), 0, stream,
                     alpha, rgbs, out, n_rays);
}